// ParallelMinGRU_56487409877669
// MI455X (gfx1250) — compile-verified
//
#include <hip/hip_runtime.h>
#include <hip/hip_bf16.h>
#include <stdint.h>

// ---------------------------------------------------------------------------
// minGRU parallel scan for MI455X (gfx1250, wave32, WMMA + async-to-LDS).
//
// Phase 1: fused dual-GEMM (x@Wz^T, x@Wh^T) with V_WMMA_F32_16X16X4_F32,
//          fp32 end-to-end to match the fp32 reference. LDS tiles are filled
//          with GLOBAL_LOAD_ASYNC_TO_LDS_B128 (ASYNCcnt) and double-buffered
//          so global loads overlap WMMA compute. Epilogue emits the scan
//          coefficients a_t = sigmoid(-k), b_t = sigmoid(k)*g(pre_h).
// Phase 2: chunked linear scan h_t = a_t h_{t-1} + b_t (stable: a in (0,1)).
// Phase 3: sequential carry combine across 16 chunks (seeded with g(h_0)).
// Phase 4: fixup out += cumA * carry.
// ---------------------------------------------------------------------------

typedef __attribute__((ext_vector_type(2))) float v2f;
typedef __attribute__((ext_vector_type(8))) float v8f;

#define B_   4
#define L_   8192
#define DX_  512
#define DH_  512

// GEMM tiling: block = 256 threads = 8 waves; block tile = 128(M=l) x 64(N=h),
// K in chunks of 32, double-buffered in LDS. Padded row stride 36 floats:
// bank-conflict-free for the fragment pattern (row*36 + kb + 2*hi) and keeps
// every 16B async chunk 16B-aligned (36*4 = 144 = 9*16).
#define TM   128
#define TN   64
#define TK   32
#define LDSS 36

// CDNA5 async copy: global -> LDS, 16B per lane, tracked by ASYNCcnt.
__device__ __forceinline__ void async_b128(uint32_t lds_off, const float* gaddr) {
    asm volatile("global_load_async_to_lds_b128 %0, %1, off"
                 :: "v"(lds_off), "v"(gaddr) : "memory");
}
__device__ __forceinline__ void wait_async0() {
    asm volatile("s_wait_asynccnt 0x0" ::: "memory");
}

__global__ __launch_bounds__(256) void mingru_gemm_gate(
    const float* __restrict__ x,
    const float* __restrict__ Wz, const float* __restrict__ bz,
    const float* __restrict__ Wh, const float* __restrict__ bh,
    float* __restrict__ Aarr, float* __restrict__ Bvarr)
{
    __shared__ __attribute__((aligned(16))) float xs [2][TM * LDSS]; // x tile  [row][k]
    __shared__ __attribute__((aligned(16))) float wzs[2][TN * LDSS]; // Wz tile [n][k]
    __shared__ __attribute__((aligned(16))) float whs[2][TN * LDSS]; // Wh tile [n][k]

    const int tid  = threadIdx.x;
    const int lane = tid & 31;
    const int wave = tid >> 5;      // 0..7 -> M sub-tile
    const int m    = lane & 15;
    const int hi   = lane >> 4;     // lane half selects K pair (A/B frags)

    const int row0 = blockIdx.x * TM;   // flat row in (B*L)
    const int h0   = blockIdx.y * TN;

    // Async staging: each thread issues 8 b128 copies per K-chunk.
    const int c4 = tid & 7;         // which 16B chunk of a 32-float row
    const int rr = tid >> 3;        // 0..31

    auto prefetch = [&](int p, int k0) {
        const uint32_t xb = (uint32_t)(uintptr_t)&xs [p][0];
        const uint32_t zb = (uint32_t)(uintptr_t)&wzs[p][0];
        const uint32_t hb = (uint32_t)(uintptr_t)&whs[p][0];
        #pragma unroll
        for (int i = 0; i < 4; ++i) {               // 128 x rows
            const int row = rr + 32 * i;
            async_b128(xb + 4u * (uint32_t)(row * LDSS + 4 * c4),
                       x + (size_t)(row0 + row) * DX_ + k0 + 4 * c4);
        }
        #pragma unroll
        for (int i = 0; i < 2; ++i) {               // 64 weight rows each
            const int row = rr + 32 * i;
            async_b128(zb + 4u * (uint32_t)(row * LDSS + 4 * c4),
                       Wz + (size_t)(h0 + row) * DX_ + k0 + 4 * c4);
            async_b128(hb + 4u * (uint32_t)(row * LDSS + 4 * c4),
                       Wh + (size_t)(h0 + row) * DX_ + k0 + 4 * c4);
        }
    };

    v8f accZ[4] = {};   // 4 N-tiles for the z-gate GEMM
    v8f accH[4] = {};   // 4 N-tiles for the h-candidate GEMM

    prefetch(0, 0);

    for (int k0 = 0; k0 < DX_; k0 += TK) {
        const int p = (k0 / TK) & 1;
        wait_async0();          // this wave's async copies (buffer p) landed
        __syncthreads();        // everyone's copies visible; buffer p^1 free
        if (k0 + TK < DX_) prefetch(p ^ 1, k0 + TK);   // overlap with compute

        #pragma unroll
        for (int kb = 0; kb < TK; kb += 4) {
            // A frag 16x4 f32: lane m = row, VGPR0/1 = K {0,1} (lanes 0-15)
            // or K {2,3} (lanes 16-31)  -> one ds_load_b64 per lane.
            const float2 af = *(const float2*)(&xs[p][(wave * 16 + m) * LDSS + kb + 2 * hi]);
            v2f a; a.x = af.x; a.y = af.y;
            #pragma unroll
            for (int nt = 0; nt < 4; ++nt) {
                const float2 bzf = *(const float2*)(&wzs[p][(nt * 16 + m) * LDSS + kb + 2 * hi]);
                v2f b0; b0.x = bzf.x; b0.y = bzf.y;
                accZ[nt] = __builtin_amdgcn_wmma_f32_16x16x4_f32(
                    false, a, false, b0, (short)0, accZ[nt], false, false);
                const float2 bhf = *(const float2*)(&whs[p][(nt * 16 + m) * LDSS + kb + 2 * hi]);
                v2f b1; b1.x = bhf.x; b1.y = bhf.y;
                accH[nt] = __builtin_amdgcn_wmma_f32_16x16x4_f32(
                    false, a, false, b1, (short)0, accH[nt], false, false);
            }
        }
    }

    // ---- epilogue: gate math, emit a_t and b_t ----
    // C/D layout: acc[r] lane -> M = r + 8*hi, N = lane&15.
    #pragma unroll
    for (int nt = 0; nt < 4; ++nt) {
        const int h = h0 + nt * 16 + m;
        const float bzv = bz[h];
        const float bhv = bh[h];
        #pragma unroll
        for (int r = 0; r < 8; ++r) {
            const int row = row0 + wave * 16 + r + 8 * hi;
            const float kz = accZ[nt][r] + bzv;
            const float ph = accH[nt][r] + bhv;
            const float ag = 1.0f / (1.0f + __expf(kz));        // 1 - z = sigmoid(-k)
            const float z  = 1.0f - ag;                         // sigmoid(k)
            const float g  = (ph >= 0.0f) ? (ph + 0.5f)
                                          : (1.0f / (1.0f + __expf(-ph)));
            const size_t idx = (size_t)row * DH_ + h;
            Aarr[idx]  = ag;
            Bvarr[idx] = z * g;
        }
    }
}

// ---------------------------------------------------------------------------
#define CHUNK  512
#define NCHUNK (L_ / CHUNK)   // 16

__global__ __launch_bounds__(256) void mingru_chunk_scan(
    float* __restrict__ Aarr,          // in: a_t ; out: within-chunk cumprod(a)
    const float* __restrict__ Bvarr,
    float* __restrict__ out)           // local (carry-free) h_t
{
    const int h = blockIdx.x * blockDim.x + threadIdx.x;  // grid.x = DH/256
    const int b = blockIdx.y;
    const int c = blockIdx.z;
    size_t idx = (size_t)(b * L_ + c * CHUNK) * DH_ + h;
    float hloc = 0.0f, cA = 1.0f;
    for (int t = 0; t < CHUNK; ++t) {
        const float a  = Aarr[idx];
        const float bv = Bvarr[idx];
        hloc = fmaf(a, hloc, bv);
        cA  *= a;
        out[idx]  = hloc;
        Aarr[idx] = cA;
        idx += DH_;
    }
}

__global__ void mingru_carry(
    const float* __restrict__ Aarr, const float* __restrict__ out,
    const float* __restrict__ h0, float* __restrict__ carry)
{
    const int t = blockIdx.x * blockDim.x + threadIdx.x;
    if (t >= B_ * DH_) return;
    const int b = t / DH_;
    const int h = t % DH_;
    const float v = h0[t];
    // reference seeds the scan with g(h_0)
    float prev = (v >= 0.0f) ? (v + 0.5f) : (1.0f / (1.0f + __expf(-v)));
    for (int c = 0; c < NCHUNK; ++c) {
        carry[(size_t)c * (B_ * DH_) + t] = prev;
        const size_t last = (size_t)(b * L_ + c * CHUNK + CHUNK - 1) * DH_ + h;
        prev = fmaf(Aarr[last], prev, out[last]);
    }
}

__global__ __launch_bounds__(256) void mingru_fixup(
    const float* __restrict__ Aarr, const float* __restrict__ carry,
    float* __restrict__ out)
{
    const size_t q = (size_t)blockIdx.x * blockDim.x + threadIdx.x;
    const size_t base = q * 4;
    if (base >= (size_t)B_ * L_ * DH_) return;
    const int    h   = (int)(base % DH_);
    const size_t row = base / DH_;            // b*L + l
    const int    b   = (int)(row / L_);
    const int    l   = (int)(row % L_);
    const int    c   = l / CHUNK;
    const float4 a4 = *(const float4*)(Aarr + base);
    const float4 o4 = *(const float4*)(out + base);
    const float4 c4 = *(const float4*)(carry + (size_t)c * (B_ * DH_) + b * DH_ + h);
    float4 r;
    r.x = fmaf(a4.x, c4.x, o4.x);
    r.y = fmaf(a4.y, c4.y, o4.y);
    r.z = fmaf(a4.z, c4.z, o4.z);
    r.w = fmaf(a4.w, c4.w, o4.w);
    *(float4*)(out + base) = r;
}

// ---------------------------------------------------------------------------
extern "C" void kernel_launch(void* const* d_in, const int* in_sizes, int n_in,
                              void* d_out, int out_size, void* d_ws, size_t ws_size,
                              hipStream_t stream) {
    const float* x  = (const float*)d_in[0];
    const float* h0 = (const float*)d_in[1];
    const float* Wz = (const float*)d_in[2];
    const float* bz = (const float*)d_in[3];
    const float* Wh = (const float*)d_in[4];
    const float* bh = (const float*)d_in[5];
    float* out = (float*)d_out;

    const size_t N = (size_t)B_ * L_ * DH_;
    float* Aarr  = (float*)d_ws;       // a_t, later cumprod(a) within chunk
    float* Bvarr = Aarr + N;           // b_t
    float* carry = Bvarr + N;          // [NCHUNK][B][DH]

    dim3 g1(B_ * L_ / TM, DH_ / TN);   // 256 x 8 blocks
    mingru_gemm_gate<<<g1, 256, 0, stream>>>(x, Wz, bz, Wh, bh, Aarr, Bvarr);

    dim3 g2(DH_ / 256, B_, NCHUNK);    // 2 x 4 x 16 blocks
    mingru_chunk_scan<<<g2, 256, 0, stream>>>(Aarr, Bvarr, out);

    mingru_carry<<<(B_ * DH_ + 255) / 256, 256, 0, stream>>>(Aarr, out, h0, carry);

    const size_t nq = N / 4;
    mingru_fixup<<<(unsigned)((nq + 255) / 256), 256, 0, stream>>>(Aarr, carry, out);
}